// CFOTLayer_54348516163934
// MI455X (gfx1250) — compile-verified
//
#include <hip/hip_runtime.h>

#define BB 64
#define CC 64
#define TT 512
#define VV 25
#define EE 64
#define TM1 511
#define NCNT (BB*TM1*VV)   // 817600 columns per BN population

typedef __attribute__((ext_vector_type(2))) float v2f;
typedef __attribute__((ext_vector_type(8))) float v8f;

// workspace float offsets
#define OFF_GRAM_ALL   0
#define OFF_SUM_ALL    4096
#define OFF_GRAM_FIRST 4160
#define OFF_SUM_FIRST  8256
#define OFF_GRAM_LAST  8320
#define OFF_SUM_LAST   12416
#define OFF_AEFF       12480
#define OFF_BEFF       16576
#define OFF_BIASA      20672
#define OFF_BIASB      20736
#define OFF_GRAM_MSG   20800
#define OFF_SUM_MSG    24896
#define OFF_WEFF       24960
#define OFF_BIASP      29056
#define WS_FLOATS      29120

static __device__ inline v8f vzero8() {
  v8f z = {0.f,0.f,0.f,0.f,0.f,0.f,0.f,0.f};
  return z;
}

// V_WMMA_F32_16X16X4_F32 wrapper: D = A(16x4) * B(4x16) + C(16x16)
static __device__ inline v8f wmma4(v2f a, v2f b, v8f c) {
  return __builtin_amdgcn_wmma_f32_16x16x4_f32(
      /*neg_a=*/false, a, /*neg_b=*/false, b,
      /*c_mod=*/(short)0, c, /*reuse_a=*/false, /*reuse_b=*/false);
}

// A operand (16x4 tile of row-major mat, row stride ld, origin m0,k0)
// lanes 0-15: VGPR0=K0, VGPR1=K1 ; lanes 16-31: VGPR0=K2, VGPR1=K3
static __device__ inline v2f load_a(const float* mat, int ld, int m0, int k0, int lane) {
  int m = m0 + (lane & 15);
  int kk = k0 + ((lane >> 4) << 1);
  v2f a; a.x = mat[m*ld + kk]; a.y = mat[m*ld + kk + 1];
  return a;
}
// B operand (4x16 tile of row-major K x N mat, row stride ld, origin k0,n0)
static __device__ inline v2f load_b(const float* mat, int ld, int k0, int n0, int lane) {
  int n = n0 + (lane & 15);
  int kk = k0 + ((lane >> 4) << 1);
  v2f b; b.x = mat[kk*ld + n]; b.y = mat[(kk+1)*ld + n];
  return b;
}
// B operand where logical B[k][n] = mat[n][k]  (transposed fetch, for X*X^T)
static __device__ inline v2f load_bt(const float* mat, int ld, int k0, int n0, int lane) {
  int n = n0 + (lane & 15);
  int kk = k0 + ((lane >> 4) << 1);
  v2f b; b.x = mat[n*ld + kk]; b.y = mat[n*ld + kk + 1];
  return b;
}

// ---------------- K0: zero stats workspace ----------------
__global__ __launch_bounds__(256) void k_zero(float* ws, int n) {
  int i = blockIdx.x * 256 + threadIdx.x;
  if (i < n) ws[i] = 0.f;
}

// ---------------- K1: channel sum + Gram of x over all t (WMMA) ----------------
__global__ __launch_bounds__(256) void k_stats_x(const float* __restrict__ x,
                                                 float* __restrict__ ws) {
  __shared__ float lX[CC][28];
  int b  = blockIdx.x >> 5;
  int t0 = (blockIdx.x & 31) << 4;
  int tid = threadIdx.x, lane = tid & 31, wv = tid >> 5;
  int gm0 = (wv >> 2) * 16,       gn0 = (wv & 3) * 16;
  int gm1 = (((wv + 8) >> 2)) * 16, gn1 = ((wv + 8) & 3) * 16;
  v8f g0 = vzero8(), g1 = vzero8();
  float ssum = 0.f;

  for (int tt = 0; tt < 16; ++tt) {
    int t = t0 + tt;
    for (int idx = tid; idx < CC * 28; idx += 256) {
      int c = idx / 28, v = idx % 28;
      lX[c][v] = (v < VV) ? x[((b * CC + c) * TT + t) * VV + v] : 0.f;
    }
    __syncthreads();
#pragma unroll
    for (int ks = 0; ks < 7; ++ks) {      // K = 28 (cols 25..27 zero)
      int k0 = ks * 4;
      g0 = wmma4(load_a(&lX[0][0], 28, gm0, k0, lane),
                 load_bt(&lX[0][0], 28, k0, gn0, lane), g0);
      g1 = wmma4(load_a(&lX[0][0], 28, gm1, k0, lane),
                 load_bt(&lX[0][0], 28, k0, gn1, lane), g1);
    }
    if (tid < CC) { float s = 0.f; for (int v = 0; v < VV; ++v) s += lX[tid][v]; ssum += s; }
    __syncthreads();
  }
  int nn = lane & 15, hi = lane >> 4;
#pragma unroll
  for (int r = 0; r < 8; ++r) {
    atomicAdd(&ws[OFF_GRAM_ALL + (gm0 + r + hi * 8) * 64 + (gn0 + nn)], g0[r]);
    atomicAdd(&ws[OFF_GRAM_ALL + (gm1 + r + hi * 8) * 64 + (gn1 + nn)], g1[r]);
  }
  if (tid < CC) atomicAdd(&ws[OFF_SUM_ALL + tid], ssum);
}

// ---------------- K1b: boundary slices t=0 and t=T-1 ----------------
__global__ __launch_bounds__(256) void k_stats_edge(const float* __restrict__ x,
                                                    float* __restrict__ ws) {
  __shared__ float l0[CC][VV];
  __shared__ float l1[CC][VV];
  int b = blockIdx.x, tid = threadIdx.x;
  for (int idx = tid; idx < CC * VV; idx += 256) {
    int c = idx / VV, v = idx % VV;
    l0[c][v] = x[((b * CC + c) * TT + 0) * VV + v];
    l1[c][v] = x[((b * CC + c) * TT + (TT - 1)) * VV + v];
  }
  __syncthreads();
  for (int e = tid; e < 4096; e += 256) {
    int ci = e >> 6, cj = e & 63;
    float s0 = 0.f, s1 = 0.f;
    for (int v = 0; v < VV; ++v) { s0 += l0[ci][v] * l0[cj][v]; s1 += l1[ci][v] * l1[cj][v]; }
    atomicAdd(&ws[OFF_GRAM_FIRST + e], s0);
    atomicAdd(&ws[OFF_GRAM_LAST + e], s1);
  }
  if (tid < CC) {
    float s0 = 0.f, s1 = 0.f;
    for (int v = 0; v < VV; ++v) { s0 += l0[tid][v]; s1 += l1[tid][v]; }
    atomicAdd(&ws[OFF_SUM_FIRST + tid], s0);
    atomicAdd(&ws[OFF_SUM_LAST + tid], s1);
  }
}

// ---------------- K2: fold BN(embed) into A_eff/B_eff + biases ----------------
__global__ __launch_bounds__(256) void k_prepare(const float* __restrict__ We,
                                                 const float* __restrict__ ge,
                                                 const float* __restrict__ be,
                                                 const float* __restrict__ Wf1,
                                                 const float* __restrict__ Wf2,
                                                 float* __restrict__ ws) {
  __shared__ float sT[EE], tT[EE], sD[EE], tD[EE];
  __shared__ float mxT[CC], mxD[CC];
  int tid = threadIdx.x;
  const float invN = 1.0f / (float)NCNT;
  if (tid < CC) {
    mxT[tid] = (ws[OFF_SUM_ALL + tid] - ws[OFF_SUM_LAST + tid]) * invN;   // Xt slice mean
    mxD[tid] = (ws[OFF_SUM_ALL + tid] - ws[OFF_SUM_FIRST + tid]) * invN;  // Xtd slice mean
  }
  __syncthreads();
  if (tid < EE) {
    int o = tid;
    float meT = 0.f, meD = 0.f;
    for (int c = 0; c < CC; ++c) { float w = We[o * CC + c]; meT += w * mxT[c]; meD += w * mxD[c]; }
    float e2T = 0.f, e2D = 0.f;
    for (int c = 0; c < CC; ++c) {
      float wc = We[o * CC + c];
      float qT = 0.f, qD = 0.f;
      for (int c2 = 0; c2 < CC; ++c2) {
        float ga = ws[OFF_GRAM_ALL + c * 64 + c2];
        qT += We[o * CC + c2] * (ga - ws[OFF_GRAM_LAST + c * 64 + c2]);
        qD += We[o * CC + c2] * (ga - ws[OFF_GRAM_FIRST + c * 64 + c2]);
      }
      e2T += wc * qT; e2D += wc * qD;
    }
    e2T *= invN; e2D *= invN;
    float vT = e2T - meT * meT, vD = e2D - meD * meD;
    float ssT = ge[o] * rsqrtf(vT + 1e-5f);
    float ssD = ge[o] * rsqrtf(vD + 1e-5f);
    sT[o] = ssT; tT[o] = be[o] - ssT * meT;
    sD[o] = ssD; tD[o] = be[o] - ssD * meD;
  }
  __syncthreads();
  for (int idx = tid; idx < EE * CC; idx += 256) {
    int o = idx >> 6, c = idx & 63;
    float accA = 0.f, accB = 0.f;
    for (int e = 0; e < EE; ++e) {
      float we = We[e * CC + c];
      accA += Wf1[o * EE + e] * sT[e] * we;
      accB += Wf2[o * EE + e] * sD[e] * we;
    }
    ws[OFF_AEFF + idx] = accA;
    ws[OFF_BEFF + idx] = accB;
  }
  if (tid < EE) {
    float ba = 0.f, bb2 = 0.f;
    for (int e = 0; e < EE; ++e) { ba += Wf1[tid * EE + e] * tT[e]; bb2 += Wf2[tid * EE + e] * tD[e]; }
    ws[OFF_BIASA + tid] = ba; ws[OFF_BIASB + tid] = bb2;
  }
}

// ---------------- K3: fused features -> OT -> msg (staged in d_out) ----------------
__global__ __launch_bounds__(256) void k_main(const float* __restrict__ x,
                                              float* __restrict__ out,
                                              float* __restrict__ ws) {
  __shared__ float lA[EE][EE];       // A_eff
  __shared__ float lB[EE][EE];       // B_eff
  __shared__ float lX[CC][28];       // Xt (zero-padded K)
  __shared__ float lXd[CC][28];      // Xtd; overlaid by P (32x32) later
  __shared__ float laM[EE][32];      // a features; overlaid by msg (64x32) later
  __shared__ float lbM[EE][32];      // b features
  __shared__ float lbiasA[EE], lbiasB[EE];
  __shared__ float lvt[VV], lvtd[VV], laa[VV], lbb[VV];

  float* lP   = &lXd[0][0];   // 32x32, stride 32
  float* lmsg = &laM[0][0];   // 64x32, stride 32

  int b  = blockIdx.x >> 5;
  int t0 = (blockIdx.x & 31) << 4;
  int tcount = (t0 + 16 <= TM1) ? 16 : (TM1 - t0);
  int tid = threadIdx.x, lane = tid & 31, wv = tid >> 5;

  for (int idx = tid; idx < EE * EE; idx += 256) {
    lA[idx >> 6][idx & 63] = ws[OFF_AEFF + idx];
    lB[idx >> 6][idx & 63] = ws[OFF_BEFF + idx];
  }
  if (tid < EE) { lbiasA[tid] = ws[OFF_BIASA + tid]; lbiasB[tid] = ws[OFF_BIASB + tid]; }
  __syncthreads();

  // persistent msg-Gram accumulators: wave wv owns tiles wv and wv+8 of the 4x4 grid
  int gm0 = (wv >> 2) * 16,         gn0 = (wv & 3) * 16;
  int gm1 = (((wv + 8) >> 2)) * 16, gn1 = ((wv + 8) & 3) * 16;
  v8f g0 = vzero8(), g1 = vzero8();
  float smsum = 0.f;

  for (int tt = 0; tt < tcount; ++tt) {
    int t = t0 + tt;
    // stage Xt / Xtd columns
    for (int idx = tid; idx < CC * 28; idx += 256) {
      int c = idx / 28, v = idx % 28;
      float xv = 0.f, xdv = 0.f;
      if (v < VV) {
        const float* base = &x[((b * CC + c) * TT + t) * VV + v];
        xv = base[0]; xdv = base[VV];
      }
      lX[c][v] = xv; lXd[c][v] = xdv;
    }
    if (tt + 1 < tcount && tid < CC)
      __builtin_prefetch(&x[((b * CC + tid) * TT + t + 2) * VV], 0, 1);
    // per-vertex velocities
    if (tid < VV) {
      float s = 0.f;
      if (t > 0)
        for (int c = 0; c < 3; ++c) {
          float d = x[((b * CC + c) * TT + t) * VV + tid] - x[((b * CC + c) * TT + t - 1) * VV + tid];
          s += d * d;
        }
      lvt[tid] = (t > 0) ? sqrtf(s) : 0.f;
    } else if (tid >= 32 && tid < 32 + VV) {
      int v = tid - 32; float s = 0.f;
      for (int c = 0; c < 3; ++c) {
        float d = x[((b * CC + c) * TT + t + 1) * VV + v] - x[((b * CC + c) * TT + t) * VV + v];
        s += d * d;
      }
      lvtd[v] = sqrtf(s);
    }
    __syncthreads();

    // a = A_eff@Xt + bias_a ; b = B_eff@Xtd + bias_b   (64x32 each, 8 waves x 1 tile pair)
    {
      int m0 = (wv >> 1) * 16, n0 = (wv & 1) * 16;
      v8f ca = vzero8(), cb = vzero8();
#pragma unroll
      for (int ks = 0; ks < 16; ++ks) {   // K = 64
        int k0 = ks * 4;
        ca = wmma4(load_a(&lA[0][0], 64, m0, k0, lane),
                   load_b(&lX[0][0], 28, k0, n0, lane), ca);
        cb = wmma4(load_a(&lB[0][0], 64, m0, k0, lane),
                   load_b(&lXd[0][0], 28, k0, n0, lane), cb);
      }
      int nn = lane & 15, hi = lane >> 4;
#pragma unroll
      for (int r = 0; r < 8; ++r) {
        int m = m0 + r + hi * 8, n = n0 + nn;
        laM[m][n] = ca[r] + lbiasA[m];
        lbM[m][n] = cb[r] + lbiasB[m];
      }
    }
    __syncthreads();

    // squared norms
    if (tid < VV) {
      float s = 0.f; for (int e = 0; e < EE; ++e) { float a = laM[e][tid]; s += a * a; }
      laa[tid] = s;
    } else if (tid >= 32 && tid < 32 + VV) {
      int v = tid - 32;
      float s = 0.f; for (int e = 0; e < EE; ++e) { float a = lbM[e][v]; s += a * a; }
      lbb[v] = s;
    }
    __syncthreads();

    // cost -> K = exp(S/tau)+eps into zero-padded 32x32 P (overlays lXd)
    for (int idx = tid; idx < 32 * 32; idx += 256) {
      int v = idx >> 5, w = idx & 31;
      float val = 0.f;
      if (v < VV && w < VV) {
        float ab = 0.f;
        for (int e = 0; e < EE; ++e) ab += laM[e][v] * lbM[e][w];
        float d2 = laa[v] + lbb[w] - 2.f * ab;
        d2 = d2 > 0.f ? d2 : 0.f;
        float S = -sqrtf(d2 + 1e-8f) - 0.2f * fabsf(lvt[v] - lvtd[w]);
        S = fminf(8.f, fmaxf(-8.f, S));
        val = __expf(S * 2.0f) + 1e-6f;   // tau = 0.5
      }
      lP[idx] = val;
    }
    __syncthreads();

    // Sinkhorn: 5 x (row-normalize, col-normalize)
    for (int it = 0; it < 5; ++it) {
      if (tid < VV) {
        float s = 0.f; for (int w2 = 0; w2 < VV; ++w2) s += lP[tid * 32 + w2];
        float inv = 1.f / (s + 1e-6f);
        for (int w2 = 0; w2 < VV; ++w2) lP[tid * 32 + w2] = lP[tid * 32 + w2] * inv + 1e-6f;
      }
      __syncthreads();
      if (tid < VV) {
        float s = 0.f; for (int v2 = 0; v2 < VV; ++v2) s += lP[v2 * 32 + tid];
        float inv = 1.f / (s + 1e-6f);
        for (int v2 = 0; v2 < VV; ++v2) lP[v2 * 32 + tid] = lP[v2 * 32 + tid] * inv + 1e-6f;
      }
      __syncthreads();
    }

    // adaptive sparsify per target column (selection until mass >= 0.9, keep >=2)
    if (tid < VV) {
      int w = tid;
      unsigned used = 0, keep = 0; float csum = 0.f;
      for (int r = 0; r < VV; ++r) {
        int jb = 0; float vb = -1e30f;
        for (int j = 0; j < VV; ++j)
          if (!((used >> j) & 1u)) { float pv = lP[j * 32 + w]; if (pv > vb) { vb = pv; jb = j; } }
        used |= 1u << jb; csum += vb;
        if (csum < 0.9f || r < 2) keep |= 1u << jb;
        else break;
      }
      float ns = 0.f;
      for (int j = 0; j < VV; ++j) if ((keep >> j) & 1u) ns += lP[j * 32 + w];
      float inv = 1.f / (ns + 1e-6f);
      for (int j = 0; j < VV; ++j) {
        float pv = lP[j * 32 + w];
        lP[j * 32 + w] = ((keep >> j) & 1u) ? pv * inv : 0.f;
      }
    }
    __syncthreads();

    // msg = Xt @ P (64x32, K=28) ; stage to lmsg + global (t+1 plane of out)
    {
      int m0 = (wv >> 1) * 16, n0 = (wv & 1) * 16;
      v8f cm = vzero8();
#pragma unroll
      for (int ks = 0; ks < 7; ++ks) {
        int k0 = ks * 4;
        cm = wmma4(load_a(&lX[0][0], 28, m0, k0, lane),
                   load_b(lP, 32, k0, n0, lane), cm);
      }
      int nn = lane & 15, hi = lane >> 4;
#pragma unroll
      for (int r = 0; r < 8; ++r) {
        int m = m0 + r + hi * 8, n = n0 + nn;
        lmsg[m * 32 + n] = cm[r];
        if (n < VV) out[((b * CC + m) * TT + (t + 1)) * VV + n] = cm[r];
      }
    }
    __syncthreads();

    // accumulate msg Gram (cols 25..31 of lmsg are exactly zero) and channel sums
#pragma unroll
    for (int ks = 0; ks < 7; ++ks) {
      int k0 = ks * 4;
      g0 = wmma4(load_a(lmsg, 32, gm0, k0, lane), load_bt(lmsg, 32, k0, gn0, lane), g0);
      g1 = wmma4(load_a(lmsg, 32, gm1, k0, lane), load_bt(lmsg, 32, k0, gn1, lane), g1);
    }
    if (tid < CC) { float s = 0.f; for (int v = 0; v < VV; ++v) s += lmsg[tid * 32 + v]; smsum += s; }
    __syncthreads();
  }

  int nn = lane & 15, hi = lane >> 4;
#pragma unroll
  for (int r = 0; r < 8; ++r) {
    atomicAdd(&ws[OFF_GRAM_MSG + (gm0 + r + hi * 8) * 64 + (gn0 + nn)], g0[r]);
    atomicAdd(&ws[OFF_GRAM_MSG + (gm1 + r + hi * 8) * 64 + (gn1 + nn)], g1[r]);
  }
  if (tid < CC) atomicAdd(&ws[OFF_SUM_MSG + tid], smsum);
}

// ---------------- K4: fold BN(proj) into W_eff + bias ----------------
__global__ __launch_bounds__(256) void k_finalize(const float* __restrict__ Wp,
                                                  const float* __restrict__ gp,
                                                  const float* __restrict__ bp,
                                                  float* __restrict__ ws) {
  __shared__ float mm[CC];
  __shared__ float sp[CC];
  int tid = threadIdx.x;
  const float invN = 1.f / (float)NCNT;
  if (tid < CC) mm[tid] = ws[OFF_SUM_MSG + tid] * invN;
  __syncthreads();
  if (tid < CC) {
    int o = tid; float my = 0.f;
    for (int c = 0; c < CC; ++c) my += Wp[o * CC + c] * mm[c];
    float e2 = 0.f;
    for (int c = 0; c < CC; ++c) {
      float q = 0.f;
      for (int c2 = 0; c2 < CC; ++c2) q += Wp[o * CC + c2] * ws[OFF_GRAM_MSG + c * 64 + c2];
      e2 += Wp[o * CC + c] * q;
    }
    e2 *= invN;
    float var = e2 - my * my;
    float s = gp[o] * rsqrtf(var + 1e-5f);
    sp[o] = s;
    ws[OFF_BIASP + o] = bp[o] - s * my;
  }
  __syncthreads();
  for (int idx = tid; idx < CC * CC; idx += 256)
    ws[OFF_WEFF + idx] = sp[idx >> 6] * Wp[idx];
}

// ---------------- K5: in-place y = W_eff @ msg + bias ; zero t=0 plane ----------------
__global__ __launch_bounds__(256) void k_proj(float* __restrict__ out,
                                              const float* __restrict__ ws) {
  __shared__ float lW[CC][CC];
  __shared__ float lM[CC][28];
  __shared__ float lbias[CC];
  int b  = blockIdx.x >> 5;
  int t0 = (blockIdx.x & 31) << 4;
  int tcount = (t0 + 16 <= TM1) ? 16 : (TM1 - t0);
  int tid = threadIdx.x, lane = tid & 31, wv = tid >> 5;
  for (int idx = tid; idx < CC * CC; idx += 256) lW[idx >> 6][idx & 63] = ws[OFF_WEFF + idx];
  if (tid < CC) lbias[tid] = ws[OFF_BIASP + tid];
  if (t0 == 0) {
    for (int idx = tid; idx < CC * VV; idx += 256) {
      int c = idx / VV, v = idx % VV;
      out[((b * CC + c) * TT + 0) * VV + v] = 0.f;
    }
  }
  __syncthreads();
  int m0 = (wv >> 1) * 16, n0 = (wv & 1) * 16;
  for (int tt = 0; tt < tcount; ++tt) {
    int t = t0 + tt;
    for (int idx = tid; idx < CC * 28; idx += 256) {
      int c = idx / 28, v = idx % 28;
      lM[c][v] = (v < VV) ? out[((b * CC + c) * TT + (t + 1)) * VV + v] : 0.f;
    }
    __syncthreads();
    v8f cy = vzero8();
#pragma unroll
    for (int ks = 0; ks < 16; ++ks) {
      int k0 = ks * 4;
      cy = wmma4(load_a(&lW[0][0], 64, m0, k0, lane),
                 load_b(&lM[0][0], 28, k0, n0, lane), cy);
    }
    int nn = lane & 15, hi = lane >> 4;
#pragma unroll
    for (int r = 0; r < 8; ++r) {
      int m = m0 + r + hi * 8, n = n0 + nn;
      if (n < VV) out[((b * CC + m) * TT + (t + 1)) * VV + n] = cy[r] + lbias[m];
    }
    __syncthreads();
  }
}

extern "C" void kernel_launch(void* const* d_in, const int* in_sizes, int n_in,
                              void* d_out, int out_size, void* d_ws, size_t ws_size,
                              hipStream_t stream) {
  (void)in_sizes; (void)n_in; (void)out_size; (void)ws_size;
  const float* x   = (const float*)d_in[0];
  const float* We  = (const float*)d_in[1];
  const float* ge  = (const float*)d_in[2];
  const float* be  = (const float*)d_in[3];
  const float* Wf1 = (const float*)d_in[4];
  const float* Wf2 = (const float*)d_in[5];
  const float* Wp  = (const float*)d_in[6];
  const float* gp  = (const float*)d_in[7];
  const float* bp  = (const float*)d_in[8];
  float* out = (float*)d_out;
  float* ws  = (float*)d_ws;

  k_zero<<<(WS_FLOATS + 255) / 256, 256, 0, stream>>>(ws, WS_FLOATS);
  k_stats_x<<<2048, 256, 0, stream>>>(x, ws);
  k_stats_edge<<<64, 256, 0, stream>>>(x, ws);
  k_prepare<<<1, 256, 0, stream>>>(We, ge, be, Wf1, Wf2, ws);
  k_main<<<2048, 256, 0, stream>>>(x, out, ws);
  k_finalize<<<1, 256, 0, stream>>>(Wp, gp, bp, ws);
  k_proj<<<2048, 256, 0, stream>>>(out, ws);
}